// SparseMLP_7619271983254
// MI455X (gfx1250) — compile-verified
//
#include <hip/hip_runtime.h>

// SparseMLP on MI455X (gfx1250): 3x fused GEMM+bias(+ReLU), exact-FP32 matrix
// cores (V_WMMA_F32_16X16X4_F32), double-buffered LDS fed by CDNA5 async
// global->LDS loads (ASYNCcnt) when the toolchain exposes them.

typedef __attribute__((ext_vector_type(2))) float v2f;
typedef __attribute__((ext_vector_type(8))) float v8f;
typedef __attribute__((ext_vector_type(4))) int   v4i;

#define MLP_B 8192
#define MLP_D 4096

#if defined(__gfx1250__) && __has_builtin(__builtin_amdgcn_global_load_async_to_lds_b128)
#define USE_ASYNC_LDS 1
#else
#define USE_ASYNC_LDS 0
#endif

__device__ __forceinline__ v8f wmma4(v2f a, v2f b, v8f c) {
    return __builtin_amdgcn_wmma_f32_16x16x4_f32(
        false, a, false, b, (short)0, c, false, false);
}

__device__ __forceinline__ void wait_async0() {
#if __has_builtin(__builtin_amdgcn_s_wait_asynccnt)
    __builtin_amdgcn_s_wait_asynccnt(0);
#else
    asm volatile("s_wait_asynccnt 0" ::: "memory");
#endif
}

// Block tile: 128 (M) x 128 (N), K-tile 32. 256 threads = 8 waves in a
// 4 (M) x 2 (N) grid; each wave owns a 32x64 tile = 2x4 WMMA accumulators.
template <bool RELU>
__global__ __launch_bounds__(256)
void mlp_gemm_xwt(const float* __restrict__ X,   // [M, K] row-major
                  const float* __restrict__ W,   // [N, K] row-major (y = X*W^T)
                  const float* __restrict__ bias,// [N]
                  float* __restrict__ Y,         // [M, N]
                  int M, int N, int K)
{
    __shared__ float Xs[2][128][36];   // padded stride -> conflict-free b64 frags
    __shared__ float Ws[2][128][36];

    const int tid   = threadIdx.x;
    const int wave  = tid >> 5;
    const int lane  = tid & 31;
    const int idx   = lane & 15;        // m/n within a 16x16 frag
    const int khalf = lane >> 4;        // lanes 16..31 hold K+2,K+3 (A/B), M=r+8 (C)
    const int wm    = (wave & 3) << 5;  // wave M offset: 0,32,64,96
    const int wn    = (wave >> 2) << 6; // wave N offset: 0,64

    const int m0 = blockIdx.y * 128;
    const int n0 = blockIdx.x * 128;

    // cooperative-load slots: row = lr + 32*it, col = lc (float4 granules)
    const int lr = tid >> 3;            // 0..31
    const int lc = (tid & 7) << 2;      // 0,4,...,28

    v8f acc[2][4];
    #pragma unroll
    for (int i = 0; i < 2; ++i)
        #pragma unroll
        for (int j = 0; j < 4; ++j)
            acc[i][j] = (v8f)(0.0f);

    const int NT = K >> 5;              // K-tiles of 32

#if USE_ASYNC_LDS
    // ---- async global->LDS issue for one 128x32 tile pair ----
    auto issue_tile = [&](int kt, int b) {
        #pragma unroll
        for (int it = 0; it < 4; ++it) {
            const int r = lr + (it << 5);
            __builtin_amdgcn_global_load_async_to_lds_b128(
                (v4i*)&X[(size_t)(m0 + r) * (size_t)K + (size_t)(kt + lc)],
                (v4i*)&Xs[b][r][lc], 0, 0);
            __builtin_amdgcn_global_load_async_to_lds_b128(
                (v4i*)&W[(size_t)(n0 + r) * (size_t)K + (size_t)(kt + lc)],
                (v4i*)&Ws[b][r][lc], 0, 0);
        }
    };
#else
    float4 stX[4], stW[4];
    auto load_regs = [&](int kt) {
        #pragma unroll
        for (int it = 0; it < 4; ++it) {
            const int r = lr + (it << 5);
            stX[it] = *reinterpret_cast<const float4*>(
                &X[(size_t)(m0 + r) * (size_t)K + (size_t)(kt + lc)]);
            stW[it] = *reinterpret_cast<const float4*>(
                &W[(size_t)(n0 + r) * (size_t)K + (size_t)(kt + lc)]);
        }
    };
    auto commit_regs = [&](int b) {
        #pragma unroll
        for (int it = 0; it < 4; ++it) {
            const int r = lr + (it << 5);
            *reinterpret_cast<float4*>(&Xs[b][r][lc]) = stX[it];
            *reinterpret_cast<float4*>(&Ws[b][r][lc]) = stW[it];
        }
    };
#endif

    // ---- one K-tile of compute from buffer c: 8 steps x 8 WMMAs ----
    auto compute = [&](int c) {
        #pragma unroll
        for (int kk = 0; kk < 32; kk += 4) {
            const int ka = kk + (khalf << 1);
            v2f a0 = *reinterpret_cast<const v2f*>(&Xs[c][wm + idx][ka]);
            v2f a1 = *reinterpret_cast<const v2f*>(&Xs[c][wm + 16 + idx][ka]);
            v2f b0 = *reinterpret_cast<const v2f*>(&Ws[c][wn + idx][ka]);
            v2f b1 = *reinterpret_cast<const v2f*>(&Ws[c][wn + 16 + idx][ka]);
            v2f b2 = *reinterpret_cast<const v2f*>(&Ws[c][wn + 32 + idx][ka]);
            v2f b3 = *reinterpret_cast<const v2f*>(&Ws[c][wn + 48 + idx][ka]);
            acc[0][0] = wmma4(a0, b0, acc[0][0]);
            acc[0][1] = wmma4(a0, b1, acc[0][1]);
            acc[0][2] = wmma4(a0, b2, acc[0][2]);
            acc[0][3] = wmma4(a0, b3, acc[0][3]);
            acc[1][0] = wmma4(a1, b0, acc[1][0]);
            acc[1][1] = wmma4(a1, b1, acc[1][1]);
            acc[1][2] = wmma4(a1, b2, acc[1][2]);
            acc[1][3] = wmma4(a1, b3, acc[1][3]);
        }
    };

#if USE_ASYNC_LDS
    issue_tile(0, 0);
    for (int t = 0; t < NT; ++t) {
        const int c = t & 1;
        wait_async0();              // my tile-t loads have landed in LDS
        __syncthreads();            // all waves: tile t visible, buf p free
        if (t + 1 < NT) issue_tile((t + 1) << 5, c ^ 1);  // overlaps compute
        compute(c);
    }
#else
    load_regs(0);
    for (int t = 0; t < NT; ++t) {
        const int c = t & 1;
        commit_regs(c);                       // stores tile t (waits its loads)
        if (t + 1 < NT) load_regs((t + 1) << 5);
        __syncthreads();                      // publish buf c
        compute(c);
        __syncthreads();                      // buf c free for t+2's commit
    }
#endif

    // ---- epilogue: bias + optional ReLU (C layout: rows r / r+8 per lane half) ----
    const int rhi = khalf << 3;
    #pragma unroll
    for (int j = 0; j < 4; ++j) {
        const int col = n0 + wn + (j << 4) + idx;
        const float bv = bias[col];
        #pragma unroll
        for (int mi = 0; mi < 2; ++mi) {
            #pragma unroll
            for (int r = 0; r < 8; ++r) {
                const int row = m0 + wm + (mi << 4) + r + rhi;
                float v = acc[mi][j][r] + bv;
                if (RELU) v = fmaxf(v, 0.0f);
                Y[(size_t)row * (size_t)N + (size_t)col] = v;
            }
        }
    }
}

extern "C" void kernel_launch(void* const* d_in, const int* in_sizes, int n_in,
                              void* d_out, int out_size, void* d_ws, size_t ws_size,
                              hipStream_t stream)
{
    (void)in_sizes; (void)n_in; (void)out_size; (void)ws_size;

    const float* x  = (const float*)d_in[0];
    const float* W0 = (const float*)d_in[1];
    const float* b0 = (const float*)d_in[2];
    const float* W1 = (const float*)d_in[3];
    const float* b1 = (const float*)d_in[4];
    const float* W2 = (const float*)d_in[5];
    const float* b2 = (const float*)d_in[6];

    float* out = (float*)d_out;                 // also used as h1 temp
    float* h2  = (float*)d_ws;                  // 8192*4096*4 = 128 MB scratch

    dim3 grid(MLP_D / 128, MLP_B / 128);        // 32 x 64 blocks
    dim3 block(256);

    mlp_gemm_xwt<true ><<<grid, block, 0, stream>>>(x,   W0, b0, out, MLP_B, MLP_D, MLP_D);
    mlp_gemm_xwt<true ><<<grid, block, 0, stream>>>(out, W1, b1, h2,  MLP_B, MLP_D, MLP_D);
    mlp_gemm_xwt<false><<<grid, block, 0, stream>>>(h2,  W2, b2, out, MLP_B, MLP_D, MLP_D);
}